// ChamferLoss_21715354649628
// MI455X (gfx1250) — compile-verified
//
#include <hip/hip_runtime.h>

typedef float v2f __attribute__((ext_vector_type(2)));
typedef float v8f __attribute__((ext_vector_type(8)));

#define NPTS  4096
#define NB    8
#define TILES (NPTS / 16)   // 256 column tiles

// For each row i of X[b]: rowmin[b*NPTS + i] = min_j ||X[b,i] - Y[b,j]||^2
// One wave owns a 16-row strip; workgroup = 8 waves = 128 rows.
// P-tile computed in a single V_WMMA_F32_16X16X4_F32:
//   A[m][k] = {x0,x1,x2,1},  B[k][n] = {-2y0,-2y1,-2y2,ry},  C[m][n] = rx_m
__global__ __launch_bounds__(256) void chamfer_rowmin_kernel(
    const float* __restrict__ X, const float* __restrict__ Y,
    float* __restrict__ rowmin)
{
    __shared__ float4 sy[NPTS];                 // 64 KB: staged Y points
    const int b    = blockIdx.y;
    const int tid  = threadIdx.x;
    const int lane = tid & 31;
    const int wave = tid >> 5;

    const float* Xb = X + (size_t)b * NPTS * 3;
    const float* Yb = Y + (size_t)b * NPTS * 3;

    // Stage all Y points for this batch: {-2y0, -2y1, -2y2, ||y||^2}
    for (int p = tid; p < NPTS; p += 256) {
        float y0 = Yb[p * 3 + 0];
        float y1 = Yb[p * 3 + 1];
        float y2 = Yb[p * 3 + 2];
        float ry = y0 * y0 + y1 * y1 + y2 * y2;
        sy[p] = make_float4(-2.0f * y0, -2.0f * y1, -2.0f * y2, ry);
    }
    __syncthreads();

    const int itile = blockIdx.x * 8 + wave;    // this wave's 16-row strip
    const int i0    = itile * 16;
    const int m     = lane & 15;                // A: lane indexes M
    const int khalf = (lane >> 4) << 1;         // K chunk: 0 (lanes 0-15) or 2 (lanes 16-31)

    // A-matrix: row m = [x0, x1, x2, 1]; this lane holds K = khalf, khalf+1
    float gx = Xb[(i0 + m) * 3 + 0];
    float gy = Xb[(i0 + m) * 3 + 1];
    float gz = Xb[(i0 + m) * 3 + 2];
    float comp[4] = { gx, gy, gz, 1.0f };
    v2f a;
    a.x = comp[khalf + 0];
    a.y = comp[khalf + 1];

    // C-matrix: C VGPR v holds row (v + 8*half) -> preload rx there
    v8f c;
    const int rbase = i0 + ((lane >> 4) << 3);
    #pragma unroll
    for (int v = 0; v < 8; ++v) {
        float x0 = Xb[(rbase + v) * 3 + 0];
        float x1 = Xb[(rbase + v) * 3 + 1];
        float x2 = Xb[(rbase + v) * 3 + 2];
        c[v] = x0 * x0 + x1 * x1 + x2 * x2;
    }

    v8f rmin;
    #pragma unroll
    for (int v = 0; v < 8; ++v) rmin[v] = __builtin_inff();

    const float* syf = (const float*)sy;
    const int    n   = lane & 15;               // B: lane indexes N

    #pragma unroll 4
    for (int jt = 0; jt < TILES; ++jt) {
        // B-matrix column n = point (jt*16 + n); lane holds K = khalf, khalf+1
        const int p = jt * 16 + n;
        v2f bmat = *(const v2f*)(syf + p * 4 + khalf);   // 8B aligned ds_load_b64

        // D = A*B + C  ==  full squared-distance tile
        v8f d = __builtin_amdgcn_wmma_f32_16x16x4_f32(
            /*neg_a=*/false, a, /*neg_b=*/false, bmat,
            /*c_mod=*/(short)0, c, /*reuse_a=*/false, /*reuse_b=*/false);

        #pragma unroll
        for (int v = 0; v < 8; ++v) rmin[v] = fminf(rmin[v], d[v]);
    }

    // Row-min: reduce across the 16 lanes of each half (xor bits 0..3 stay in-half)
    #pragma unroll
    for (int v = 0; v < 8; ++v) {
        float val = rmin[v];
        val = fminf(val, __shfl_xor(val, 1, 32));
        val = fminf(val, __shfl_xor(val, 2, 32));
        val = fminf(val, __shfl_xor(val, 4, 32));
        val = fminf(val, __shfl_xor(val, 8, 32));
        rmin[v] = val;
    }
    // lane 0 holds rows i0..i0+7, lane 16 holds rows i0+8..i0+15
    if ((lane & 15) == 0) {
        const int row = i0 + ((lane >> 4) << 3);
        #pragma unroll
        for (int v = 0; v < 8; ++v)
            rowmin[(size_t)b * NPTS + row + v] = rmin[v];
    }
}

// Deterministic single-block sum of all row-mins -> scalar loss
__global__ __launch_bounds__(256) void chamfer_sum_kernel(
    const float* __restrict__ vals, int n, float* __restrict__ out)
{
    __shared__ float s[256];
    float acc = 0.0f;
    for (int i = threadIdx.x; i < n; i += 256) acc += vals[i];
    s[threadIdx.x] = acc;
    __syncthreads();
    for (int off = 128; off > 0; off >>= 1) {
        if ((int)threadIdx.x < off) s[threadIdx.x] += s[threadIdx.x + off];
        __syncthreads();
    }
    if (threadIdx.x == 0) out[0] = s[0];
}

extern "C" void kernel_launch(void* const* d_in, const int* in_sizes, int n_in,
                              void* d_out, int out_size, void* d_ws, size_t ws_size,
                              hipStream_t stream) {
    const float* preds = (const float*)d_in[0];   // [8, 4096, 3]
    const float* gts   = (const float*)d_in[1];   // [8, 4096, 3]
    float*       out   = (float*)d_out;           // scalar

    float* rm1 = (float*)d_ws;                    // [B*NPTS]: per-pred min over gts (loss_1)
    float* rm2 = rm1 + (size_t)NB * NPTS;         // [B*NPTS]: per-gt  min over preds (loss_2)

    dim3 grid(NPTS / 128, NB);                    // 32 row-blocks x 8 batches
    dim3 block(256);

    // loss_1: rows = preds, candidates = gts
    chamfer_rowmin_kernel<<<grid, block, 0, stream>>>(preds, gts, rm1);
    // loss_2: rows = gts, candidates = preds (role-symmetric second pass)
    chamfer_rowmin_kernel<<<grid, block, 0, stream>>>(gts, preds, rm2);

    // rm1 and rm2 are contiguous: one reduction over both
    chamfer_sum_kernel<<<1, block, 0, stream>>>(rm1, 2 * NB * NPTS, out);
}